// CustomMeshRasterizer_43327630082319
// MI455X (gfx1250) — compile-verified
//
#include <hip/hip_runtime.h>

#define IMG    128
#define NPIX   (IMG * IMG)
#define FOCALC 2.0f
#define EPSC   1e-8f
#define BIGC   1e10f
#define NVERT  1500
#define NFACE  2000          // 125 * 16 exactly
#define NTILE  (NFACE / 16)

typedef __attribute__((ext_vector_type(2))) float v2f;
typedef __attribute__((ext_vector_type(8))) float v8f;

// ---------------------------------------------------------------------------
// Kernel 0: verts -> NDC  (v_view = verts @ R + T ; vndc = (F*x/zs, F*y/zs, z))
// ---------------------------------------------------------------------------
__global__ void vndc_kernel(const float* __restrict__ verts,
                            const float* __restrict__ R,
                            const float* __restrict__ T,
                            float* __restrict__ vndc) {
    int v = blockIdx.x * blockDim.x + threadIdx.x;
    if (v >= NVERT) return;
    float x = verts[3 * v + 0];
    float y = verts[3 * v + 1];
    float z = verts[3 * v + 2];
    // v_view[j] = sum_i verts[i] * R[i][j] + T[j]
    float vx = x * R[0] + y * R[3] + z * R[6] + T[0];
    float vy = x * R[1] + y * R[4] + z * R[7] + T[1];
    float vz = x * R[2] + y * R[5] + z * R[8] + T[2];
    float zs = (fabsf(vz) < EPSC) ? EPSC : vz;
    float izs = __builtin_amdgcn_rcpf(zs);
    vndc[4 * v + 0] = FOCALC * vx * izs;
    vndc[4 * v + 1] = FOCALC * vy * izs;
    vndc[4 * v + 2] = vz;
    vndc[4 * v + 3] = 0.0f;
}

// ---------------------------------------------------------------------------
// Kernel 1: per-face records (16 floats):
//   [0..5]  Ax0,Ay0, Ax1,Ay1, Ax2,Ay2   (pre-scaled by 1/(area_s*zv_k))
//   [6..8]  C0,C1,C2                     (pre-scaled)
//   [9..14] ax,ay,bx,by,cx,cy            (2D NDC geometry for seg-dist)
//   [15]    pad
// Degenerate face: coeffs -> (0,0,-1) so iw_k == -1 => never inside.
// ---------------------------------------------------------------------------
__global__ void face_prep_kernel(const float* __restrict__ vndc,
                                 const int* __restrict__ faces,
                                 float* __restrict__ recs) {
    int f = blockIdx.x * blockDim.x + threadIdx.x;
    if (f >= NFACE) return;
    int i0 = faces[3 * f + 0], i1 = faces[3 * f + 1], i2 = faces[3 * f + 2];
    float ax = vndc[4 * i0], ay = vndc[4 * i0 + 1], az = vndc[4 * i0 + 2];
    float bx = vndc[4 * i1], by = vndc[4 * i1 + 1], bz = vndc[4 * i1 + 2];
    float cx = vndc[4 * i2], cy = vndc[4 * i2 + 1], cz = vndc[4 * i2 + 2];

    float area   = (bx - ax) * (cy - ay) - (by - ay) * (cx - ax);
    float area_s = (fabsf(area) < EPSC) ? EPSC : area;
    bool  valid  = fabsf(area) > EPSC;
    float z0 = fmaxf(az, EPSC), z1 = fmaxf(bz, EPSC), z2 = fmaxf(cz, EPSC);
    float s0 = 1.0f / (area_s * z0);
    float s1 = 1.0f / (area_s * z1);
    float s2 = 1.0f / (area_s * z2);

    // edge(u,v,p) = (v.x-u.x)(p.y-u.y) - (v.y-u.y)(p.x-u.x)
    //            = -(v.y-u.y)*px + (v.x-u.x)*py + [(v.y-u.y)*u.x - (v.x-u.x)*u.y]
    // edge0: (b,c), edge1: (c,a), edge2: (a,b)
    float e0x = -(cy - by) * s0, e0y = (cx - bx) * s0;
    float e0c = ((cy - by) * bx - (cx - bx) * by) * s0;
    float e1x = -(ay - cy) * s1, e1y = (ax - cx) * s1;
    float e1c = ((ay - cy) * cx - (ax - cx) * cy) * s1;
    float e2x = -(by - ay) * s2, e2y = (bx - ax) * s2;
    float e2c = ((by - ay) * ax - (bx - ax) * ay) * s2;
    if (!valid) {
        e0x = e0y = e1x = e1y = e2x = e2y = 0.0f;
        e0c = e1c = e2c = -1.0f;
    }
    float* r = recs + (size_t)f * 16;
    r[0] = e0x; r[1] = e0y; r[2] = e1x; r[3] = e1y; r[4] = e2x; r[5] = e2y;
    r[6] = e0c; r[7] = e1c; r[8] = e2c;
    r[9] = ax;  r[10] = ay; r[11] = bx; r[12] = by; r[13] = cx; r[14] = cy;
    r[15] = 0.0f;
}

// ---------------------------------------------------------------------------
// Kernel 2: rasterize. One wave32 = 16 pixels; tiles of 16 faces via
// V_WMMA_F32_16X16X4_F32 (D = Acoef(16x4) * Bpix(4x16)).
// ---------------------------------------------------------------------------
__device__ __forceinline__ float seg_d2(float ax, float ay, float bx, float by,
                                        float px, float py) {
    float abx = bx - ax, aby = by - ay;
    float t = ((px - ax) * abx + (py - ay) * aby) *
              __builtin_amdgcn_rcpf(fmaxf(abx * abx + aby * aby, EPSC));
    t = fminf(fmaxf(t, 0.0f), 1.0f);
    float dx = px - (ax + t * abx);
    float dy = py - (ay + t * aby);
    return dx * dx + dy * dy;
}

__global__ void raster_kernel(const float* __restrict__ recs,
                              float* __restrict__ out) {
    const int tid   = threadIdx.x;
    const int lane  = tid & 31;
    const bool lo   = lane < 16;
    const int  n    = lane & 15;                       // pixel column / A face row
    const int  wave = blockIdx.x * (blockDim.x >> 5) + (tid >> 5);
    const int  pixBase = wave * 16;
    const int  p    = pixBase + n;
    const int  row  = p >> 7;                          // p / 128
    const int  col  = p & 127;                         // p % 128
    const float px  = 1.0f - 2.0f * ((float)col + 0.5f) / (float)IMG;
    const float py  = 1.0f - 2.0f * ((float)row + 0.5f) / (float)IMG;

    // B matrix (4x16): K rows = (px, py, 1, 0).
    // VGPR0: lanes0-15 K=0 (px), lanes16-31 K=2 (1); VGPR1: K=1 (py) / K=3 (0)
    v2f bv;
    bv.x = lo ? px : 1.0f;
    bv.y = lo ? py : 0.0f;

    const int h8 = (lane >> 4) << 3;                   // 0 or 8 (D half select)
    float bestZ = BIGC;
    int   bestF = 0;

    for (int t = 0; t < NTILE; ++t) {
        const float* rec = recs + (size_t)(t * 16 + n) * 16;
        // speculative prefetch of next tile's records (dropped if OOB)
        __builtin_prefetch(rec + 256, 0, 1);

        // A matrices (16x4): VGPR0 = K0(Ax)/K2(C), VGPR1 = K1(Ay)/K3(0)
        v2f a0, a1, a2;
        a0.x = lo ? rec[0] : rec[6];  a0.y = lo ? rec[1] : 0.0f;
        a1.x = lo ? rec[2] : rec[7];  a1.y = lo ? rec[3] : 0.0f;
        a2.x = lo ? rec[4] : rec[8];  a2.y = lo ? rec[5] : 0.0f;

        v8f cz = {};
        v8f d0 = __builtin_amdgcn_wmma_f32_16x16x4_f32(
            false, a0, false, bv, (short)0, cz, false, false);
        v8f d1 = __builtin_amdgcn_wmma_f32_16x16x4_f32(
            false, a1, false, bv, (short)0, cz, false, false);
        v8f d2 = __builtin_amdgcn_wmma_f32_16x16x4_f32(
            false, a2, false, bv, (short)0, cz, false, false);

#pragma unroll
        for (int r = 0; r < 8; ++r) {
            float iw0 = d0[r], iw1 = d1[r], iw2 = d2[r];
            float mn    = fminf(fminf(iw0, iw1), iw2);         // v_min3
            float denom = fmaxf((iw0 + iw1) + iw2, EPSC);
            float zpix  = __builtin_amdgcn_rcpf(denom);        // single v_rcp_f32
            float z     = (mn >= 0.0f) ? zpix : BIGC;
            if (z < bestZ) { bestZ = z; bestF = t * 16 + r + h8; }
        }
    }

    // merge the two lane-halves (face rows 0-7 vs 8-15), tie-break on index
    float zO = __shfl_xor(bestZ, 16, 32);
    int   fO = __shfl_xor(bestF, 16, 32);
    if (zO < bestZ || (zO == bestZ && fO < bestF)) { bestZ = zO; bestF = fO; }

    if (lo) {
        bool hit = bestZ < (BIGC * 0.5f);
        int*   p2f  = (int*)out;
        float* zbuf = out + NPIX;
        float* bary = out + 2 * NPIX;
        float* dst  = out + 5 * NPIX;

        float b0 = -1.0f, b1 = -1.0f, b2 = -1.0f, dd = -1.0f;
        if (hit) {
            const float* rec = recs + (size_t)bestF * 16;
            b0 = (rec[0] * px + rec[1] * py + rec[6]) * bestZ;
            b1 = (rec[2] * px + rec[3] * py + rec[7]) * bestZ;
            b2 = (rec[4] * px + rec[5] * py + rec[8]) * bestZ;
            float ax = rec[9],  ay = rec[10];
            float bx = rec[11], by = rec[12];
            float cx = rec[13], cy = rec[14];
            float d2 = fminf(fminf(seg_d2(ax, ay, bx, by, px, py),
                                   seg_d2(bx, by, cx, cy, px, py)),
                             seg_d2(cx, cy, ax, ay, px, py));
            dd = -d2;           // winner is inside => signed dist is -d2
        }
        p2f[p]  = hit ? bestF : -1;
        zbuf[p] = hit ? bestZ : -1.0f;
        bary[p * 3 + 0] = b0;
        bary[p * 3 + 1] = b1;
        bary[p * 3 + 2] = b2;
        dst[p]  = dd;
    }
}

// ---------------------------------------------------------------------------
// Launch
// ---------------------------------------------------------------------------
extern "C" void kernel_launch(void* const* d_in, const int* in_sizes, int n_in,
                              void* d_out, int out_size, void* d_ws, size_t ws_size,
                              hipStream_t stream) {
    const float* verts = (const float*)d_in[0];   // 1*1500*3
    const float* R     = (const float*)d_in[1];   // 1*3*3
    const float* T     = (const float*)d_in[2];   // 1*3
    const int*   faces = (const int*)d_in[3];     // 2000*3

    float* wsf  = (float*)d_ws;
    float* vndc = wsf;                 // 1500 * 4 floats
    float* recs = wsf + 8192;          // 2000 * 16 floats (32KB offset)
    float* out  = (float*)d_out;

    vndc_kernel<<<(NVERT + 255) / 256, 256, 0, stream>>>(verts, R, T, vndc);
    face_prep_kernel<<<(NFACE + 255) / 256, 256, 0, stream>>>(vndc, faces, recs);
    // 1 wave -> 16 pixels; 256-thread block (8 waves) -> 128 pixels; 128 blocks
    raster_kernel<<<NPIX / 128, 256, 0, stream>>>(recs, out);
}